// ResourceGraphEncoder_58823872086653
// MI455X (gfx1250) — compile-verified
//
#include <hip/hip_runtime.h>
#include <hip/hip_bf16.h>

typedef __attribute__((ext_vector_type(2))) float v2f;
typedef __attribute__((ext_vector_type(8))) float v8f;

#define HIDF 128
#define KKF  256
#define GROWS 48     // rows per GEMM block (3 strips of 16)
#define LDA   260    // padded LDS stride in floats (16B-aligned rows, spreads banks)

// ---------------------------------------------------------------- utilities
__global__ void zero_kernel(float* __restrict__ p, long n) {
  long i = (long)blockIdx.x * blockDim.x + threadIdx.x;
  if (i < n) p[i] = 0.0f;
}

// Build B in WMMA-fragment-native layout:
// for kp = k0/4, hi in {0,1}, col in [0,128):
//   WF2[(kp*2 + hi)*128 + col] = ( W[4kp+2hi][col], W[4kp+2hi+1][col] )
// where W = [W_l ; W_r] stacked (256 x 128, row-major source).
// Each lane's B fragment is then one contiguous 8-byte load; a half-wave
// reads 128 contiguous bytes.
__global__ void build_wfrag_kernel(const float* __restrict__ wl,
                                   const float* __restrict__ wr,
                                   float2* __restrict__ wf) {
  int i = blockIdx.x * blockDim.x + threadIdx.x;   // 0..16383 (float2 entries)
  if (i >= 64 * 2 * HIDF) return;
  const int kp  = i >> 8;          // k0 / 4
  const int rem = i & 255;
  const int hi  = rem >> 7;
  const int col = rem & 127;
  const int k   = 4 * kp + 2 * hi; // k and k+1 are in the same half (k even)
  float a, b;
  if (k < HIDF) {
    a = wl[(size_t)k * HIDF + col];
    b = wl[(size_t)(k + 1) * HIDF + col];
  } else {
    a = wr[(size_t)(k - HIDF) * HIDF + col];
    b = wr[(size_t)(k + 1 - HIDF) * HIDF + col];
  }
  wf[i] = make_float2(a, b);
}

// ---------------------------------------------------------------- degree
__global__ void deg_count_kernel(const long long* __restrict__ dst,
                                 float* __restrict__ deg, int E) {
  int e = blockIdx.x * blockDim.x + threadIdx.x;
  if (e < E) atomicAdd(&deg[(int)dst[e]], 1.0f);
}

__global__ void deg_inv_kernel(float* __restrict__ deg, int n) {
  int i = blockIdx.x * blockDim.x + threadIdx.x;
  if (i < n) deg[i] = 1.0f / fmaxf(deg[i], 1.0f);
}

// ---------------------------------------------------------------- edge scatter
// one wave (32 lanes) per edge; each lane moves 4 consecutive features.
// x / h arrays (51 MB) are L2-resident; atomics resolve at L2 (SCOPE_DEV).
__global__ void scatter_add_kernel(const float* __restrict__ feat,
                                   const long long* __restrict__ src,
                                   const long long* __restrict__ dst,
                                   float* __restrict__ agg, int E) {
  long gid = (long)blockIdx.x * blockDim.x + threadIdx.x;
  int e = (int)(gid >> 5);
  if (e >= E) return;
  int lane = (int)(gid & 31);
  int s = (int)src[e];
  int d = (int)dst[e];
  int f = lane * 4;
  const float4 v = *(const float4*)&feat[(size_t)s * HIDF + f];
  float* p = &agg[(size_t)d * HIDF + f];
  atomicAdd(p + 0, v.x);
  atomicAdd(p + 1, v.y);
  atomicAdd(p + 2, v.z);
  atomicAdd(p + 3, v.w);
}

// ---------------------------------------------------------------- fused GEMM
// z = [agg*deg_inv | hin] @ [W_l ; W_r] + b   (K = 256, N = 128)
// also accumulates per-feature sum / sumsq for batch-norm statistics.
// 48 rows per block; 8 waves, each owns a 16-column tile and 3 row strips.
__global__ __launch_bounds__(256)
void sage_gemm_wmma(const float* __restrict__ agg,
                    const float* __restrict__ hin,
                    const float* __restrict__ dinv,
                    const float2* __restrict__ wfrag,  // fragment-layout weights
                    const float* __restrict__ bias,    // [128]
                    float* __restrict__ z,             // [M][128] (may alias hin)
                    float* __restrict__ stats,         // [2][128] sum, sumsq
                    int M) {
  __shared__ float As[GROWS * LDA];
  const int t = threadIdx.x;
  const int row0 = blockIdx.x * GROWS;

  // stage effective A rows into LDS (float4 granularity; zero-fill past M)
  for (int i = t; i < GROWS * (KKF / 4); i += 256) {
    const int r  = i / (KKF / 4);
    const int c4 = (i % (KKF / 4)) * 4;
    const int gr = row0 + r;
    float4 v = make_float4(0.f, 0.f, 0.f, 0.f);
    if (gr < M) {
      if (c4 < HIDF) {
        float4 a = *(const float4*)&agg[(size_t)gr * HIDF + c4];
        const float sc = dinv[gr];
        v = make_float4(a.x * sc, a.y * sc, a.z * sc, a.w * sc);
      } else {
        v = *(const float4*)&hin[(size_t)gr * HIDF + (c4 - HIDF)];
      }
    }
    *(float4*)&As[r * LDA + c4] = v;
  }
  __syncthreads();

  const int lane = t & 31;
  const int ln   = lane & 15;
  const int hi   = lane >> 4;        // half-wave selects K sub-pair
  const int cb   = (t >> 5) * 16;    // wave -> 16-column tile
  const int col  = cb + ln;

  // per-lane base into fragment-layout weights: advances by 256 float2 per k0+=4
  const float2* wfp = wfrag + (size_t)hi * HIDF + col;

  v8f acc[3] = {};

  for (int k0 = 0; k0 < KKF; k0 += 4) {
    // B fragment (4x16 f32) as one coalesced 8-byte load per lane
    const float2 bw = wfp[(size_t)(k0 >> 2) * (2 * HIDF)];
    v2f b;  b.x = bw.x;  b.y = bw.y;
    const int ao = k0 + 2 * hi;
#pragma unroll
    for (int s = 0; s < 3; ++s) {
      // A fragment (16x4 f32): lanes 0-15 -> M=ln K=k0..k0+1 ; lanes 16-31 -> K=k0+2..k0+3
      v2f a = *(const v2f*)&As[(s * 16 + ln) * LDA + ao];
      acc[s] = __builtin_amdgcn_wmma_f32_16x16x4_f32(
          false, a, false, b, (short)0, acc[s], false, false);
    }
  }

  // epilogue: bias, store z, BN partial stats (C layout: VGPR i -> row hi*8+i, col=ln)
  const float bv = bias[col];
  float ssum = 0.f, ssq = 0.f;
#pragma unroll
  for (int s = 0; s < 3; ++s) {
    const int rbase = row0 + s * 16 + hi * 8;
    if (row0 + s * 16 < M) {   // M % 16 == 0 -> strip-granular validity
#pragma unroll
      for (int i = 0; i < 8; ++i) {
        float v = acc[s][i] + bv;
        z[(size_t)(rbase + i) * HIDF + col] = v;
        ssum += v;
        ssq  += v * v;
      }
    }
  }
  atomicAdd(&stats[col], ssum);
  atomicAdd(&stats[HIDF + col], ssq);
}

// ---------------------------------------------------------------- batch-norm
__global__ void bn_finalize_kernel(const float* __restrict__ stats,
                                   const float* __restrict__ gamma,
                                   const float* __restrict__ beta,
                                   float* __restrict__ ss, float inv_n) {
  int c = threadIdx.x;
  if (c < HIDF) {
    float mu  = stats[c] * inv_n;
    float var = stats[HIDF + c] * inv_n - mu * mu;
    float sc  = rsqrtf(var + 1e-5f) * gamma[c];
    ss[c] = sc;
    ss[HIDF + c] = beta[c] - mu * sc;
  }
}

__global__ void bn_relu_kernel(float* __restrict__ h,
                               const float* __restrict__ ss, long n) {
  long i = ((long)blockIdx.x * blockDim.x + threadIdx.x) * 4;
  if (i >= n) return;
  int c = (int)(i & (HIDF - 1));
  float4 v = *(float4*)&h[i];
  v.x = fmaxf(0.f, v.x * ss[c + 0] + ss[HIDF + c + 0]);
  v.y = fmaxf(0.f, v.y * ss[c + 1] + ss[HIDF + c + 1]);
  v.z = fmaxf(0.f, v.z * ss[c + 2] + ss[HIDF + c + 2]);
  v.w = fmaxf(0.f, v.w * ss[c + 3] + ss[HIDF + c + 3]);
  *(float4*)&h[i] = v;
}

// layer-2 BN+ReLU fused with column-wise max reduction into out (128 floats)
__global__ void bn_relu_max_kernel(const float* __restrict__ z,
                                   const float* __restrict__ ss,
                                   float* __restrict__ out, int M) {
  __shared__ float red[256];
  const int col  = threadIdx.x & (HIDF - 1);
  const int half = threadIdx.x >> 7;
  const float sc = ss[col];
  const float sh = ss[HIDF + col];
  float m = 0.0f;   // ReLU output is >= 0
  for (long r = (long)blockIdx.x * 2 + half; r < M; r += (long)gridDim.x * 2) {
    float v = fmaxf(0.f, z[r * HIDF + col] * sc + sh);
    m = fmaxf(m, v);
  }
  red[threadIdx.x] = m;
  __syncthreads();
  if (threadIdx.x < HIDF) {
    m = fmaxf(red[threadIdx.x], red[threadIdx.x + HIDF]);
    // values are non-negative: int bit pattern preserves float order
    atomicMax((int*)&out[threadIdx.x], __float_as_int(m));
  }
}

// ---------------------------------------------------------------- launch
extern "C" void kernel_launch(void* const* d_in, const int* in_sizes, int n_in,
                              void* d_out, int out_size, void* d_ws, size_t ws_size,
                              hipStream_t stream) {
  const float*     x    = (const float*)d_in[0];
  const long long* ei   = (const long long*)d_in[1];   // (2, E) int64
  const float*     W1l  = (const float*)d_in[2];
  const float*     b1l  = (const float*)d_in[3];
  const float*     W1r  = (const float*)d_in[4];
  const float*     W2l  = (const float*)d_in[5];
  const float*     b2l  = (const float*)d_in[6];
  const float*     W2r  = (const float*)d_in[7];
  const float*     g1   = (const float*)d_in[8];
  const float*     be1  = (const float*)d_in[9];
  const float*     g2   = (const float*)d_in[10];
  const float*     be2  = (const float*)d_in[11];
  float*           out  = (float*)d_out;

  const int N = in_sizes[0] / HIDF;     // 100000
  const int E = in_sizes[1] / 2;        // 1600000
  const long NH = (long)N * HIDF;

  const long long* esrc = ei;
  const long long* edst = ei + E;

  // workspace carve-up (floats)
  float* DEG   = (float*)d_ws;               // N (deg -> deg_inv in place)
  float* AGG   = DEG + 102400;               // N*128
  float* H     = AGG + NH;                   // N*128 (z / h, in place)
  float* WC1   = H + NH;                     // 256*128 (fragment layout)
  float* WC2   = WC1 + 32768;                // 256*128 (fragment layout)
  float* STATS = WC2 + 32768;                // 256
  float* SS    = STATS + 256;                // 256

  const int B = 256;
  const int gemm_blocks = (N + GROWS - 1) / GROWS;
  auto blocks = [](long n, int b) { return (int)((n + b - 1) / b); };

  // ---- init
  zero_kernel<<<blocks(N, B), B, 0, stream>>>(DEG, N);
  zero_kernel<<<blocks(NH, B), B, 0, stream>>>(AGG, NH);
  zero_kernel<<<1, B, 0, stream>>>(STATS, 256);
  zero_kernel<<<1, B, 0, stream>>>(out, HIDF);
  build_wfrag_kernel<<<64, B, 0, stream>>>(W1l, W1r, (float2*)WC1);
  build_wfrag_kernel<<<64, B, 0, stream>>>(W2l, W2r, (float2*)WC2);

  // ---- degrees
  deg_count_kernel<<<blocks(E, B), B, 0, stream>>>(edst, DEG, E);
  deg_inv_kernel<<<blocks(N, B), B, 0, stream>>>(DEG, N);

  // ---- layer 1
  scatter_add_kernel<<<blocks((long)E * 32, B), B, 0, stream>>>(x, esrc, edst, AGG, E);
  sage_gemm_wmma<<<gemm_blocks, B, 0, stream>>>(AGG, x, DEG, (const float2*)WC1, b1l,
                                                H, STATS, N);
  bn_finalize_kernel<<<1, 128, 0, stream>>>(STATS, g1, be1, SS, 1.0f / (float)N);
  bn_relu_kernel<<<blocks(NH / 4, B), B, 0, stream>>>(H, SS, NH);

  // ---- layer 2
  zero_kernel<<<blocks(NH, B), B, 0, stream>>>(AGG, NH);
  zero_kernel<<<1, B, 0, stream>>>(STATS, 256);
  scatter_add_kernel<<<blocks((long)E * 32, B), B, 0, stream>>>(H, esrc, edst, AGG, E);
  sage_gemm_wmma<<<gemm_blocks, B, 0, stream>>>(AGG, H, DEG, (const float2*)WC2, b2l,
                                                H, STATS, N);
  bn_finalize_kernel<<<1, 128, 0, stream>>>(STATS, g2, be2, SS, 1.0f / (float)N);

  // ---- graph readout: max over nodes
  bn_relu_max_kernel<<<256, B, 0, stream>>>(H, SS, out, N);
}